// Instrument2_17832704213487
// MI455X (gfx1250) — compile-verified
//
#include <hip/hip_runtime.h>
#include <math.h>

typedef float v2f __attribute__((ext_vector_type(2)));
typedef float v8f __attribute__((ext_vector_type(8)));

#define PI_D 3.14159265358979323846

// ---------------------------------------------------------------------------
// Kernel 1: closed-form fft_shift kernels K(d; s), d in [-2047,2047].
// Rows 0..15: input shifts (per be). Rows 16..79: frame shifts f/63.
// K(d) = [sin(3071.5*a)/sin(a/2) + cos(3072*a)]/6144, a = 2*pi*d/6144 - theta
// theta = 2*pi*(s*2048/3)/3073.  Evaluated in double.
// Frame rows additionally stored REVERSED (Krev[j] = K[4094-j]) so the WMMA
// B-fragments {K[bd], K[bd-1]} become one ascending (mergeable) b64 load.
// ---------------------------------------------------------------------------
__global__ void k_tables(const float* __restrict__ shifts,
                         float* __restrict__ Ktab,
                         float* __restrict__ Krev) {
    int row = blockIdx.x;  // 0..79
    double s = (row < 16) ? (double)shifts[row] : (double)(row - 16) / 63.0;
    double shift_samples = s * (2048.0 / 3.0);
    double theta = 2.0 * PI_D * shift_samples / 3073.0;
    for (int j = threadIdx.x; j < 4095; j += blockDim.x) {
        int d = j - 2047;
        double a  = 2.0 * PI_D * (double)d / 6144.0 - theta;
        double sh = sin(0.5 * a);
        double D  = (fabs(sh) < 1e-12) ? 6143.0 : (sin(3071.5 * a) / sh);
        double S  = D + cos(3072.0 * a);
        float val = (float)(S / 6144.0);
        Ktab[row * 4096 + j] = val;
        if (row >= 16) Krev[(row - 16) * 4096 + (4094 - j)] = val;
    }
    if (threadIdx.x == 0) {
        Ktab[row * 4096 + 4095] = 0.0f;
        if (row >= 16) Krev[(row - 16) * 4096 + 4095] = 0.0f;
    }
}

// ---------------------------------------------------------------------------
// Kernel 2: t1[be][t] = sum_m time[be][m] * Kshift_be[t-m+2047]
// grid (8,16), block 256
// ---------------------------------------------------------------------------
__global__ void k_shift_time(const float* __restrict__ timein,
                             const float* __restrict__ Ktab,
                             float* __restrict__ t1) {
    __shared__ float ts[2048];
    __shared__ float ks[4096];
    int be = blockIdx.y;
    for (int i = threadIdx.x; i < 2048; i += 256) ts[i] = timein[be * 2048 + i];
    for (int i = threadIdx.x; i < 4096; i += 256) ks[i] = Ktab[be * 4096 + i];
    __syncthreads();
    int t = blockIdx.x * 256 + threadIdx.x;
    int base = t + 2047;
    float acc = 0.0f;
    for (int m = 0; m < 2048; ++m) acc += ts[m] * ks[base - m];
    t1[be * 2048 + t] = acc;
}

// ---------------------------------------------------------------------------
// Kernel 3: exp_t[be][f][t] = sum_m t1[be][m] * Kframe_f[t-m+2047]
// f32 WMMA 16x16x4: A = t1 (16 be x 2048 m) staged in LDS; B from the
// reversed Toeplitz tables. Each wave handles TWO frame rows (f0, f0+1)
// sharing one A-fragment: per step 1 ds_load + 2 global b64 + 2 WMMAs.
// 32 f-pairs * 128 t-tiles = 4096 wave-tiles; 8 waves/block -> 512 blocks.
// ---------------------------------------------------------------------------
__global__ void k_expt_wmma(const float* __restrict__ t1,
                            const float* __restrict__ Krev,
                            float* __restrict__ expt) {
    __shared__ float As[16][516];
    int wave = threadIdx.x >> 5;
    int lane = threadIdx.x & 31;
    int tile = blockIdx.x * 8 + wave;
    int f0 = (tile >> 7) << 1;           // 0,2,...,62
    int t0 = (tile & 127) << 4;
    const float* Kr0 = Krev + f0 * 4096;
    const float* Kr1 = Kr0 + 4096;

    int row  = lane & 15;        // A-row (be)
    int hi   = lane >> 4;        // half-wave
    int koff = hi << 1;          // A k-offset within 4-chunk: {0,2}
    int col  = lane & 15;        // B/D column (t)
    int krow = hi << 1;          // B k-row within 4-chunk: {0,2}

    v8f c0 = {0.f, 0.f, 0.f, 0.f, 0.f, 0.f, 0.f, 0.f};
    v8f c1 = {0.f, 0.f, 0.f, 0.f, 0.f, 0.f, 0.f, 0.f};

    int rbase = 2047 - t0 - col + krow;  // + m0 gives reversed index

    for (int ch = 0; ch < 4; ++ch) {
        int mbase = ch * 512;
        __syncthreads();
        for (int i = threadIdx.x; i < 16 * 512; i += 256) {
            int r = i >> 9, mm = i & 511;
            As[r][mm] = t1[r * 2048 + mbase + mm];
        }
        __syncthreads();
        for (int k = 0; k < 128; ++k) {
            int mloc = k << 2;
            v2f a;
            a.x = As[row][mloc + koff];
            a.y = As[row][mloc + koff + 1];
            int r = rbase + mbase + mloc;       // Krev index, ascending pair
            v2f b0, b1;
            b0.x = Kr0[r];
            b0.y = Kr0[r + 1];
            b1.x = Kr1[r];
            b1.y = Kr1[r + 1];
            c0 = __builtin_amdgcn_wmma_f32_16x16x4_f32(false, a, false, b0,
                                                       (short)0, c0, false, false);
            c1 = __builtin_amdgcn_wmma_f32_16x16x4_f32(false, a, false, b1,
                                                       (short)0, c1, false, false);
        }
    }
    int t = t0 + col;
    #pragma unroll
    for (int j = 0; j < 8; ++j) {
        int be = j + (hi << 3);                 // D: VGPR j -> M=j / j+8
        expt[(be * 64 + f0) * 2048 + t]       = c0[j];
        expt[(be * 64 + f0 + 1) * 2048 + t]   = c1[j];
    }
}

// ---------------------------------------------------------------------------
// Kernel 4: shifted step-mask = suffix sum of the shift kernel:
//   msk[be][f][t] = sum_{m=f+2}^{2047} Kshift_be[t-m+2047]
// Backward scan over m, emitting at m = f+2 (f = 0..63). grid (8,16), block 256
// ---------------------------------------------------------------------------
__global__ void k_mask(const float* __restrict__ Ktab, float* __restrict__ msk) {
    __shared__ float ks[4096];
    int be = blockIdx.y;
    for (int i = threadIdx.x; i < 4096; i += 256) ks[i] = Ktab[be * 4096 + i];
    __syncthreads();
    int t = blockIdx.x * 256 + threadIdx.x;
    float acc = 0.0f;
    for (int m = 2047; m >= 2; --m) {
        acc += ks[t - m + 2047];
        if (m <= 65) msk[(be * 64 + (m - 2)) * 2048 + t] = acc;
    }
}

// ---------------------------------------------------------------------------
// Kernel 5: env[be][cp][t] = sum_f energy[be,cp,f]*exp(-k*exp_t[be,f,t])*msk[be,f,t]
// k = friction/(2*mass). grid (512, 8), block 256 (t fastest -> coalesced).
// ---------------------------------------------------------------------------
__global__ void k_env(const float* __restrict__ energy,
                      const float* __restrict__ props,
                      const float* __restrict__ expt,
                      const float* __restrict__ msk,
                      float* __restrict__ env) {
    __shared__ float en[64];
    __shared__ float kk;
    int becp = blockIdx.x;
    int be = becp >> 5;
    if (threadIdx.x < 64) en[threadIdx.x] = energy[becp * 64 + threadIdx.x];
    if (threadIdx.x == 0) {
        float mass = props[becp * 2 + 0];
        float fric = props[becp * 2 + 1];
        kk = fric / (2.0f * mass);
    }
    __syncthreads();
    int t = blockIdx.y * 256 + threadIdx.x;
    const float* E = expt + be * 64 * 2048 + t;
    const float* M = msk  + be * 64 * 2048 + t;
    float k = kk;
    float acc = 0.0f;
    #pragma unroll 4
    for (int f = 0; f < 64; ++f)
        acc += en[f] * expf(-k * E[f * 2048]) * M[f * 2048];
    env[becp * 2048 + t] = acc;
}

// ---------------------------------------------------------------------------
// Kernel 6: residual MLP on shape (tiny): s[f][c] (64x32 per be), 4 layers of
//   s = s + relu(s) @ W^T + b, final relu, stored transposed as sfin[be][c][f].
// grid 16, block 256.
// ---------------------------------------------------------------------------
__global__ void k_mlp(const float* __restrict__ shapein,
                      const float* __restrict__ wl,
                      const float* __restrict__ bl,
                      float* __restrict__ sfin) {
    __shared__ float s0[2048], s1[2048], W[1024], Bv[32];
    int be = blockIdx.x;
    for (int i = threadIdx.x; i < 2048; i += 256) {
        int f = i >> 5, c = i & 31;
        s0[i] = shapein[(be * 32 + c) * 64 + f];
    }
    float* cur = s0;
    float* nxt = s1;
    for (int L = 0; L < 4; ++L) {
        __syncthreads();
        for (int i = threadIdx.x; i < 1024; i += 256) W[i] = wl[L * 1024 + i];
        if (threadIdx.x < 32) Bv[threadIdx.x] = bl[L * 32 + threadIdx.x];
        __syncthreads();
        for (int i = threadIdx.x; i < 2048; i += 256) {
            int f = i >> 5, c = i & 31;
            float acc = cur[i] + Bv[c];
            #pragma unroll
            for (int j = 0; j < 32; ++j) {
                float r = cur[(f << 5) + j];
                r = r > 0.0f ? r : 0.0f;
                acc += r * W[c * 32 + j];
            }
            nxt[i] = acc;
        }
        float* tmp = cur; cur = nxt; nxt = tmp;
    }
    __syncthreads();
    for (int i = threadIdx.x; i < 2048; i += 256) {
        int f = i >> 5, c = i & 31;
        float v = cur[i];
        v = v > 0.0f ? v : 0.0f;
        sfin[(be * 32 + c) * 64 + f] = v;
    }
}

// ---------------------------------------------------------------------------
// Kernel 7: pos[ch][T] = sum_e sin(tt[T]*freqs[e]*pi) * w_pos[ch][e] + b_pos[ch]
// f32 WMMA 16x16x4: each wave computes BOTH 16-channel M-tiles against one
// shared sine B-fragment (2 sinf amortized over 2 WMMAs). 2048 t-tiles,
// 8 waves/block -> 256 blocks.
// ---------------------------------------------------------------------------
__global__ void k_pos_wmma(const float* __restrict__ wpos,
                           const float* __restrict__ bpos,
                           float* __restrict__ pos) {
    __shared__ float Ws[32][260];
    int wave = threadIdx.x >> 5;
    int lane = threadIdx.x & 31;
    for (int i = threadIdx.x; i < 32 * 256; i += 256) {
        int r = i >> 8, e = i & 255;
        Ws[r][e] = wpos[r * 256 + e];
    }
    __syncthreads();
    int tile = blockIdx.x * 8 + wave;         // 2048 t-tiles
    int t0   = tile << 4;

    int row  = lane & 15;
    int hi   = lane >> 4;
    int koff = hi << 1;
    int col  = lane & 15;
    int krow = hi << 1;

    const float FSTEP = (0.5f - 1e-5f) / 255.0f;
    float tcol = (float)(t0 + col) * (32768.0f / 32767.0f);   // tt[T]

    v8f c0 = {0.f, 0.f, 0.f, 0.f, 0.f, 0.f, 0.f, 0.f};
    v8f c1 = {0.f, 0.f, 0.f, 0.f, 0.f, 0.f, 0.f, 0.f};
    for (int k = 0; k < 64; ++k) {
        int e0 = k << 2;
        float f0 = 1e-5f + (float)(e0 + krow)     * FSTEP;
        float f1 = 1e-5f + (float)(e0 + krow + 1) * FSTEP;
        v2f b;
        b.x = sinf(tcol * f0 * 3.14159265358979323846f);
        b.y = sinf(tcol * f1 * 3.14159265358979323846f);
        v2f a0, a1;
        a0.x = Ws[row][e0 + koff];
        a0.y = Ws[row][e0 + koff + 1];
        a1.x = Ws[16 + row][e0 + koff];
        a1.y = Ws[16 + row][e0 + koff + 1];
        c0 = __builtin_amdgcn_wmma_f32_16x16x4_f32(false, a0, false, b,
                                                   (short)0, c0, false, false);
        c1 = __builtin_amdgcn_wmma_f32_16x16x4_f32(false, a1, false, b,
                                                   (short)0, c1, false, false);
    }
    int t = t0 + col;
    #pragma unroll
    for (int j = 0; j < 8; ++j) {
        int ch = j + (hi << 3);
        pos[ch * 32768 + t]            = c0[j] + bpos[ch];
        pos[(16 + ch) * 32768 + t]     = c1[j] + bpos[16 + ch];
    }
}

// ---------------------------------------------------------------------------
// Kernel 8: fused interpolation + final reduction:
//   out[be][T] = sum_ch pos[ch][T] * interp64(sfin[be][ch]) *
//                relu(interp2048(env[be][ch]))
// grid 2048, block 256.
// ---------------------------------------------------------------------------
__global__ void k_final(const float* __restrict__ pos,
                        const float* __restrict__ sfin,
                        const float* __restrict__ env,
                        float* __restrict__ out) {
    int idx = blockIdx.x * 256 + threadIdx.x;   // be*32768 + T
    int be = idx >> 15;
    int T  = idx & 32767;

    float ce = ((float)T + 0.5f) * 0.0625f - 0.5f;        // 2048 -> 32768
    ce = fminf(fmaxf(ce, 0.0f), 2047.0f);
    int i0 = (int)floorf(ce);
    int i1 = min(i0 + 1, 2047);
    float w = ce - (float)i0;

    float cs = ((float)T + 0.5f) * (1.0f / 512.0f) - 0.5f; // 64 -> 32768
    cs = fminf(fmaxf(cs, 0.0f), 63.0f);
    int j0 = (int)floorf(cs);
    int j1 = min(j0 + 1, 63);
    float v = cs - (float)j0;

    float acc = 0.0f;
    #pragma unroll 4
    for (int ch = 0; ch < 32; ++ch) {
        const float* er = env + (be * 32 + ch) * 2048;
        float e = er[i0] * (1.0f - w) + er[i1] * w;
        e = fmaxf(e, 0.0f);
        const float* sr = sfin + (be * 32 + ch) * 64;
        float sv = sr[j0] * (1.0f - v) + sr[j1] * v;
        acc += pos[ch * 32768 + T] * sv * e;
    }
    out[idx] = acc;
}

// ---------------------------------------------------------------------------
extern "C" void kernel_launch(void* const* d_in, const int* in_sizes, int n_in,
                              void* d_out, int out_size, void* d_ws, size_t ws_size,
                              hipStream_t stream) {
    (void)in_sizes; (void)n_in; (void)out_size; (void)ws_size;
    const float* timein = (const float*)d_in[0];  // (2,8,2048)
    const float* shifts = (const float*)d_in[1];  // (2,8,1)
    const float* energy = (const float*)d_in[2];  // (2,8,32,64)
    const float* shapein= (const float*)d_in[3];  // (2,8,32,64)
    const float* props  = (const float*)d_in[4];  // (2,8,32,2)
    const float* wpos   = (const float*)d_in[5];  // (32,256)
    const float* bpos   = (const float*)d_in[6];  // (32,)
    const float* wl     = (const float*)d_in[7];  // (4,32,32)
    const float* bl     = (const float*)d_in[8];  // (4,32)
    float* out = (float*)d_out;                   // (2,8,32768)

    float* ws     = (float*)d_ws;
    float* Ktab   = ws;                    // 80*4096
    float* Krev   = Ktab + 80 * 4096;      // 64*4096 (reversed frame tables)
    float* t1     = Krev + 64 * 4096;      // 16*2048
    float* expt   = t1   + 16 * 2048;      // 16*64*2048
    float* msk    = expt + 16 * 64 * 2048; // 16*64*2048
    float* env    = msk  + 16 * 64 * 2048; // 16*32*2048
    float* sfin   = env  + 16 * 32 * 2048; // 16*32*64
    float* pos    = sfin + 16 * 32 * 64;   // 32*32768
    // total = 6,946,816 floats = ~27.8 MB of workspace

    k_tables    <<<80, 256, 0, stream>>>(shifts, Ktab, Krev);
    k_shift_time<<<dim3(8, 16), 256, 0, stream>>>(timein, Ktab, t1);
    k_expt_wmma <<<512, 256, 0, stream>>>(t1, Krev, expt);
    k_mask      <<<dim3(8, 16), 256, 0, stream>>>(Ktab, msk);
    k_env       <<<dim3(512, 8), 256, 0, stream>>>(energy, props, expt, msk, env);
    k_mlp       <<<16, 256, 0, stream>>>(shapein, wl, bl, sfin);
    k_pos_wmma  <<<256, 256, 0, stream>>>(wpos, bpos, pos);
    k_final     <<<2048, 256, 0, stream>>>(pos, sfin, env, out);
}